// SSIM_55602646614035
// MI455X (gfx1250) — compile-verified
//
#include <hip/hip_runtime.h>
#include <math.h>

// ---------------------------------------------------------------------------
// SSIM on gfx1250 (MI455X): separable 11-tap Gaussian convs expressed as
// banded matmuls on V_WMMA_F32_16X16X4_F32 (full fp32 tensor path).
// One wave32 = one 16x16 output tile. 8 waves per workgroup.
//  - clamp+select boundary handling (no exec-mask branches)
//  - p / y loaded once per tile; p^2, y^2, p*y formed in registers
//  - wave-private LDS staging ordered with s_wait_dscnt (no __syncthreads)
//  - SSIM map fused to a single v_rcp division + raw v_sqrt per element
// ---------------------------------------------------------------------------

typedef __attribute__((ext_vector_type(2))) float v2f;
typedef __attribute__((ext_vector_type(8))) float v8f;

#define IMG_H 768
#define IMG_W 768
#define N_IMG 32
#define TILES_X (IMG_W / 16)                    // 48
#define TILES_PER_IMG (TILES_X * (IMG_H / 16))  // 2304
#define N_TILES (N_IMG * TILES_PER_IMG)         // 73728
#define WAVES_PER_BLOCK 8
#define N_BLOCKS (N_TILES / WAVES_PER_BLOCK)    // 9216
#define LDS_STRIDE 34                           // even -> 8B-aligned b64 reads

#define SSIM_C1 1.0e-4f       // 0.01^2
#define SSIM_C2 9.0e-4f       // 0.03^2

// 11-tap Gaussian (sigma = 1.5, hardcoded by reference), normalized.
// g[k] = exp(-(k-5)^2/4.5)/Z = exp2(d^2 * (-1/(4.5*ln2)) + log2(1/Z)), d=k-5.
__device__ __forceinline__ float gw(int k) {
  float d = (float)(k - 5);
  float v = __builtin_amdgcn_exp2f(fmaf(d * d, -0.32059845f, -1.9104355f));
  return ((unsigned)k < 11u) ? v : 0.0f;
}

// Wave-local ordering for the wave-private LDS staging buffer.
// LDS ops of one wave are processed in order; this waits for completion and
// acts as a compiler barrier so DS reads/writes are not reordered.
__device__ __forceinline__ void wave_ds_fence() {
  asm volatile("s_wait_dscnt 0x0" ::: "memory");
}

__device__ __forceinline__ v8f wmma_f32(v2f a, v2f b, v8f c) {
  return __builtin_amdgcn_wmma_f32_16x16x4_f32(false, a, false, b, (short)0, c,
                                               false, false);
}

// Full 2-D conv of one channel; raw[c][4] = {r0@colA, r1@colA, r0@colB, r1@colB}
// composed from preloaded p/y values by the caller-selected CH.
template <int CH>
__device__ __forceinline__ v8f conv_channel(const float pr[7][4],
                                            const float yr[7][4],
                                            int n, int half, const v2f band[7],
                                            float* __restrict__ st) {
  v8f acc0 = {0.f, 0.f, 0.f, 0.f, 0.f, 0.f, 0.f, 0.f};
  v8f acc1 = {0.f, 0.f, 0.f, 0.f, 0.f, 0.f, 0.f, 0.f};
#pragma unroll
  for (int c = 0; c < 7; ++c) {
    float v0, v1, v2, v3;
    if (CH == 0) {
      v0 = pr[c][0]; v1 = pr[c][1]; v2 = pr[c][2]; v3 = pr[c][3];
    } else if (CH == 1) {
      v0 = yr[c][0]; v1 = yr[c][1]; v2 = yr[c][2]; v3 = yr[c][3];
    } else if (CH == 2) {
      v0 = pr[c][0] * pr[c][0]; v1 = pr[c][1] * pr[c][1];
      v2 = pr[c][2] * pr[c][2]; v3 = pr[c][3] * pr[c][3];
    } else if (CH == 3) {
      v0 = yr[c][0] * yr[c][0]; v1 = yr[c][1] * yr[c][1];
      v2 = yr[c][2] * yr[c][2]; v3 = yr[c][3] * yr[c][3];
    } else {
      v0 = pr[c][0] * yr[c][0]; v1 = pr[c][1] * yr[c][1];
      v2 = pr[c][2] * yr[c][2]; v3 = pr[c][3] * yr[c][3];
    }
    v2f b0 = {v0, v1};
    v2f b1 = {v2, v3};
    acc0 = wmma_f32(band[c], b0, acc0);   // vertical pass, cols 0..15
    acc1 = wmma_f32(band[c], b1, acc1);   // vertical pass, cols 16..31
  }

  // D-layout -> row-major stage (wave-private buffer)
  wave_ds_fence();                        // prior channel's reads complete
#pragma unroll
  for (int v = 0; v < 8; ++v) {
    const int m = v + 8 * half;
    st[m * LDS_STRIDE + n]      = acc0[v];
    st[m * LDS_STRIDE + 16 + n] = acc1[v];
  }
  wave_ds_fence();                        // writes visible to A-layout reads

  v8f out = {0.f, 0.f, 0.f, 0.f, 0.f, 0.f, 0.f, 0.f};
#pragma unroll
  for (int c = 0; c < 7; ++c) {
    const int j0 = 4 * c + 2 * half;      // even -> 8B-aligned b64 read
    v2f a = *(const v2f*)(st + n * LDS_STRIDE + j0);
    out = wmma_f32(a, band[c], out);      // horizontal pass
  }
  return out;
}

__global__ __launch_bounds__(256)
void ssim_tile_kernel(const float* __restrict__ pred,
                      const float* __restrict__ yref,
                      float* __restrict__ partial) {
  const int lane = threadIdx.x & 31;
  const int wave = threadIdx.x >> 5;
  const int tile = blockIdx.x * WAVES_PER_BLOCK + wave;

  const int img = tile / TILES_PER_IMG;
  const int t   = tile % TILES_PER_IMG;
  const int y0  = (t / TILES_X) * 16;
  const int x0  = (t % TILES_X) * 16;

  const float* __restrict__ P = pred + (size_t)img * (IMG_H * IMG_W);
  const float* __restrict__ Y = yref + (size_t)img * (IMG_H * IMG_W);

  const int n    = lane & 15;
  const int half = lane >> 4;

  // Band matrix per-lane values: g[j - (lane%16)], j = 4c + 2*half (+1).
  // Identical for the vertical A operand and the horizontal B operand.
  v2f band[7];
#pragma unroll
  for (int c = 0; c < 7; ++c) {
    const int j0 = 4 * c + 2 * half;
    band[c].x = gw(j0     - n);
    band[c].y = gw(j0 + 1 - n);
  }

  // ---- preload p and y halo region in B-operand layout (clamp + select) ----
  const int cAi = x0 - 5 + n;
  const int cBi = cAi + 16;
  const bool okA = (unsigned)cAi < (unsigned)IMG_W;
  const bool okB = (unsigned)cBi < (unsigned)IMG_W;
  const int colA = min(max(cAi, 0), IMG_W - 1);
  const int colB = min(max(cBi, 0), IMG_W - 1);

  float pr[7][4];
  float yr[7][4];
#pragma unroll
  for (int c = 0; c < 7; ++c) {
    const int r0 = y0 - 5 + 4 * c + 2 * half;
    const int r1 = r0 + 1;
    const bool ok0 = (unsigned)r0 < (unsigned)IMG_H;
    const bool ok1 = (unsigned)r1 < (unsigned)IMG_H;
    const unsigned b0 = (unsigned)(min(max(r0, 0), IMG_H - 1)) * IMG_W;
    const unsigned b1 = (unsigned)(min(max(r1, 0), IMG_H - 1)) * IMG_W;

    float t0 = P[b0 + colA], t1 = P[b1 + colA];
    float t2 = P[b0 + colB], t3 = P[b1 + colB];
    float u0 = Y[b0 + colA], u1 = Y[b1 + colA];
    float u2 = Y[b0 + colB], u3 = Y[b1 + colB];
    pr[c][0] = (ok0 && okA) ? t0 : 0.f;
    pr[c][1] = (ok1 && okA) ? t1 : 0.f;
    pr[c][2] = (ok0 && okB) ? t2 : 0.f;
    pr[c][3] = (ok1 && okB) ? t3 : 0.f;
    yr[c][0] = (ok0 && okA) ? u0 : 0.f;
    yr[c][1] = (ok1 && okA) ? u1 : 0.f;
    yr[c][2] = (ok0 && okB) ? u2 : 0.f;
    yr[c][3] = (ok1 && okB) ? u3 : 0.f;
  }

  __shared__ float lds[WAVES_PER_BLOCK][16 * LDS_STRIDE];  // per-wave stage
  float* st = &lds[wave][0];

  v8f mu1 = conv_channel<0>(pr, yr, n, half, band, st);
  v8f mu2 = conv_channel<1>(pr, yr, n, half, band, st);
  v8f cpp = conv_channel<2>(pr, yr, n, half, band, st);
  v8f cyy = conv_channel<3>(pr, yr, n, half, band, st);
  v8f cpy = conv_channel<4>(pr, yr, n, half, band, st);

  float s = 0.0f;
#pragma unroll
  for (int v = 0; v < 8; ++v) {
    const float m1  = mu1[v];
    const float m2  = mu2[v];
    const float m11 = m1 * m1;
    const float m22 = m2 * m2;
    const float m12 = m1 * m2;
    const float s1  = cpp[v] - m11;
    const float s2  = cyy[v] - m22;
    const float s12 = cpy[v] - m12;
    // lum*con*str fused into one quotient; raw v_sqrt / v_rcp (~1 ulp)
    const float num = (2.0f * m12 + SSIM_C1) * (2.0f * s12 + SSIM_C2) *
                      (s12 + 0.5f * SSIM_C2);
    const float rt  = __builtin_amdgcn_sqrtf(s1 * s2);
    const float den = (m11 + m22 + SSIM_C1) * (s1 + s2 + SSIM_C2) *
                      (rt + 0.5f * SSIM_C2);
    s += num * __builtin_amdgcn_rcpf(den);
  }

  // wave32 reduction
#pragma unroll
  for (int off = 16; off > 0; off >>= 1)
    s += __shfl_down(s, off, 32);

  __shared__ float wsum[WAVES_PER_BLOCK];
  if (lane == 0) wsum[wave] = s;
  __syncthreads();
  if (threadIdx.x == 0) {
    float tsum = 0.0f;
#pragma unroll
    for (int i = 0; i < WAVES_PER_BLOCK; ++i) tsum += wsum[i];
    partial[blockIdx.x] = tsum;
  }
}

__global__ __launch_bounds__(256)
void ssim_reduce_kernel(const float* __restrict__ partial, int np,
                        float* __restrict__ out, float inv_count) {
  __shared__ float sh[256];
  float s = 0.0f;
  for (int i = threadIdx.x; i < np; i += 256) s += partial[i];
  sh[threadIdx.x] = s;
  __syncthreads();
  for (int stride = 128; stride > 0; stride >>= 1) {
    if ((int)threadIdx.x < stride) sh[threadIdx.x] += sh[threadIdx.x + stride];
    __syncthreads();
  }
  if (threadIdx.x == 0) out[0] = sh[0] * inv_count;
}

extern "C" void kernel_launch(void* const* d_in, const int* in_sizes, int n_in,
                              void* d_out, int out_size, void* d_ws, size_t ws_size,
                              hipStream_t stream) {
  (void)in_sizes; (void)n_in; (void)out_size; (void)ws_size;
  const float* pred = (const float*)d_in[0];
  const float* yref = (const float*)d_in[1];
  float* out     = (float*)d_out;
  float* partial = (float*)d_ws;   // N_BLOCKS floats of scratch

  ssim_tile_kernel<<<N_BLOCKS, 256, 0, stream>>>(pred, yref, partial);
  const float inv_count = 1.0f / ((float)N_IMG * (float)IMG_H * (float)IMG_W);
  ssim_reduce_kernel<<<1, 256, 0, stream>>>(partial, N_BLOCKS, out, inv_count);
}